// ECELoss_52913997087017
// MI455X (gfx1250) — compile-verified
//
#include <hip/hip_runtime.h>
#include <hip/hip_bf16.h>

#define ECE_NBINS 20
#define ECE_C     128   // classes per row (known from reference)

// ---------------------------------------------------------------------------
// Kernel 0: zero the 3*20-word scratch accumulator in d_ws.
// ---------------------------------------------------------------------------
__global__ void ece_zero_ws(unsigned* ws) {
    int t = threadIdx.x;
    if (t < 3 * ECE_NBINS) ws[t] = 0u;   // zero bit-pattern valid for uint & float
}

// ---------------------------------------------------------------------------
// Kernel 1: streaming max/argmax + binned accumulation.
//
// Layout: 8 lanes per row, 4 rows per wave32 per iteration.
//   lane L: segment sub = L>>3 selects row (row0+sub), g = L&7 selects the
//   64B chunk; each lane loads 16 consecutive floats (4x global_load_b128).
// Private 16-way max/argmax (first-occurrence ties), then ONE 3-step
// shfl_xor butterfly (masks 4,2,1) reduces all four 8-lane segments in
// parallel. Lanes 0/8/16/24 bin their row via LDS atomics; block epilogue
// flushes 20 bins to global scratch atomically.
// ---------------------------------------------------------------------------
__global__ __launch_bounds__(256) void ece_main(const float* __restrict__ logits,
                                                const int*   __restrict__ labels,
                                                unsigned*    __restrict__ g_cnt,
                                                float*       __restrict__ g_acc,
                                                float*       __restrict__ g_conf,
                                                int n_rows) {
    __shared__ unsigned s_cnt[ECE_NBINS];
    __shared__ float    s_acc[ECE_NBINS];
    __shared__ float    s_conf[ECE_NBINS];

    const int t = threadIdx.x;
    if (t < ECE_NBINS) { s_cnt[t] = 0u; s_acc[t] = 0.0f; s_conf[t] = 0.0f; }
    __syncthreads();

    const int lane          = t & 31;          // wave32 lane id
    const int sub           = lane >> 3;       // which of the 4 rows (0..3)
    const int g             = lane & 7;        // 64B-chunk within the row
    const int waveInBlock   = t >> 5;
    const int wavesPerBlock = blockDim.x >> 5;
    const int waveId        = blockIdx.x * wavesPerBlock + waveInBlock;
    const int totalWaves    = gridDim.x * wavesPerBlock;
    const int rowStride     = totalWaves * 4;

    for (int row0 = waveId * 4; row0 < n_rows; row0 += rowStride) {
        const int row  = row0 + sub;
        const int srow = (row < n_rows) ? row : (n_rows - 1);   // clamp for safe load
        const float* rp = logits + (size_t)srow * ECE_C + (size_t)g * 16;

        // gfx1250 global_prefetch_b8 on next iteration's tile (2KB/wave covered)
        if (row0 + rowStride < n_rows) {
            __builtin_prefetch(logits + (size_t)(row0 + rowStride + sub) * ECE_C +
                               (size_t)g * 16, 0, 1);
        }

        // 16 consecutive floats = 4x global_load_b128
        const float4 a = ((const float4*)rp)[0];
        const float4 b = ((const float4*)rp)[1];
        const float4 c = ((const float4*)rp)[2];
        const float4 d = ((const float4*)rp)[3];

        const float vv[16] = { a.x, a.y, a.z, a.w,  b.x, b.y, b.z, b.w,
                               c.x, c.y, c.z, c.w,  d.x, d.y, d.z, d.w };

        // private 16-way max/argmax; strict '>' keeps the first occurrence
        float m = vv[0];
        int   j = 0;
        #pragma unroll
        for (int k = 1; k < 16; ++k) {
            if (vv[k] > m) { m = vv[k]; j = k; }
        }
        int gidx = g * 16 + j;   // column index within the row

        // segmented butterfly: masks 4,2,1 stay inside each 8-lane segment,
        // reducing all 4 rows with a single 3-step pass (ties -> lowest index)
        #pragma unroll
        for (int s = 4; s > 0; s >>= 1) {
            float om = __shfl_xor(m,    s, 32);
            int   oi = __shfl_xor(gidx, s, 32);
            if (om > m || (om == m && oi < gidx)) { m = om; gidx = oi; }
        }

        if (g == 0 && row < n_rows) {   // lanes 0,8,16,24: one per row
            // bins are (b/20, (b+1)/20]; ceil handles the closed upper edge
            int bin = (int)ceilf(m * 20.0f) - 1;
            if (bin >= 0) {
                if (bin > ECE_NBINS - 1) bin = ECE_NBINS - 1;
                const float accf = (gidx == labels[row]) ? 1.0f : 0.0f;
                atomicAdd(&s_cnt[bin],  1u);   // ds_add_u32
                atomicAdd(&s_acc[bin],  accf); // ds_add_f32
                atomicAdd(&s_conf[bin], m);    // ds_add_f32
            }
        }
    }
    __syncthreads();

    if (t < ECE_NBINS && s_cnt[t] != 0u) {
        atomicAdd(&g_cnt[t],  s_cnt[t]);       // global_atomic_add_u32
        atomicAdd(&g_acc[t],  s_acc[t]);       // global_atomic_add_f32
        atomicAdd(&g_conf[t], s_conf[t]);      // global_atomic_add_f32
    }
}

// ---------------------------------------------------------------------------
// Kernel 2: single-wave finalize. out[0] = ece, out[1..20] = acc_in.
// ---------------------------------------------------------------------------
__global__ void ece_final(const unsigned* __restrict__ g_cnt,
                          const float*    __restrict__ g_acc,
                          const float*    __restrict__ g_conf,
                          float*          __restrict__ out,
                          float inv_n) {
    const int t = threadIdx.x;    // launched with 32 threads (one wave32)
    float contrib = 0.0f;
    if (t < ECE_NBINS) {
        const float cnt     = (float)g_cnt[t];
        const float prop    = cnt * inv_n;
        const float safe    = fmaxf(cnt, 1.0f);
        const float acc_in  = (cnt > 0.0f) ? g_acc[t]  / safe : 0.0f;
        const float conf_in = (cnt > 0.0f) ? g_conf[t] / safe : 0.0f;
        contrib             = (cnt > 0.0f) ? fabsf(conf_in - acc_in) * prop : 0.0f;
        out[1 + t] = acc_in;
    }
    #pragma unroll
    for (int s = 16; s > 0; s >>= 1) contrib += __shfl_xor(contrib, s, 32);
    if (t == 0) out[0] = contrib;
}

// ---------------------------------------------------------------------------
extern "C" void kernel_launch(void* const* d_in, const int* in_sizes, int n_in,
                              void* d_out, int out_size, void* d_ws, size_t ws_size,
                              hipStream_t stream) {
    const float* logits = (const float*)d_in[0];
    const int*   labels = (const int*)d_in[1];

    const int n_rows = in_sizes[1];                 // N = 1048576 (labels count)

    unsigned* g_cnt  = (unsigned*)d_ws;             // [0 .. 19]  bin counts
    float*    g_acc  = (float*)d_ws + ECE_NBINS;    // [20 .. 39] sum(acc)
    float*    g_conf = (float*)d_ws + 2 * ECE_NBINS;// [40 .. 59] sum(conf)

    ece_zero_ws<<<1, 64, 0, stream>>>((unsigned*)d_ws);

    const int threads = 256;                        // 8 wave32s per block
    const int blocks  = 4096;                       // 32Ki waves, 4 rows/wave/iter
    ece_main<<<blocks, threads, 0, stream>>>(logits, labels,
                                             g_cnt, g_acc, g_conf, n_rows);

    ece_final<<<1, 32, 0, stream>>>(g_cnt, g_acc, g_conf,
                                    (float*)d_out, 1.0f / (float)n_rows);
}